// CalibConv_50448685859193
// MI455X (gfx1250) — compile-verified
//
#include <hip/hip_runtime.h>
#include <hip/hip_bf16.h>
#include <math.h>

// Problem constants (from reference)
#define NB   8
#define CCH  128
#define HH   56
#define WW   56
#define OCH  64
#define NC   2
#define KK   5
#define PP   (KK*KK)        // 25
#define HW   (HH*WW)        // 3136
#define NPIX (NB*HW)        // 25088
#define NCOL 50             // P*NC valid columns
#define NCOLP 64            // padded columns for WMMA tiles

// Paired-B layout: row = channel-pair cp (c = 2*cp + parity), 64 rows.
// Within a row: off = col*2 + parity  -> (B[c][col], B[c+1][col]) adjacent.
// Row stride 160 floats: 160 mod 64 banks = 32 -> consecutive rows land on
// disjoint 32-bank halves, so the two wave halves (cp vs cp+1) are
// conflict-free on ds_load_b64.
#define BROW 160
#define BROWS 64
#define BTOT (BROWS*BROW)   // 10240 floats

typedef __attribute__((ext_vector_type(2))) float v2f;
typedef __attribute__((ext_vector_type(8))) float v8f;

// ws layout (floats):
//   [0, 10240)            : Bp[64][160]  (paired folded weights)
//   [10240, 10242)        : bfold[2]
//   [10304, 10304+NPIX*50): Y[pix][50]
#define WS_BF    BTOT
#define WS_Y     (BTOT + 64)

// ---------------------------------------------------------------------------
// Kernel 1: fold Wlin into Wt  ->  paired layout
//   Bp[cp*160 + col*2 + par] = sum_o Wlin[k,o]*Wt[o, 2*cp+par, p]
//   (col = p*2+k; cols >= 50 and pad region zero-filled)
// ---------------------------------------------------------------------------
__global__ void calib_fold_kernel(const float* __restrict__ Wt,
                                  const float* __restrict__ bias,
                                  const float* __restrict__ Wlin,
                                  const float* __restrict__ blin,
                                  float* __restrict__ Bp,
                                  float* __restrict__ bfold) {
    int idx = blockIdx.x * 256 + threadIdx.x;       // 0 .. 10239
    if (idx >= BTOT) return;
    int row = idx / BROW;                           // channel pair cp
    int off = idx - row * BROW;
    float v = 0.f;
    if (off < 2 * NCOLP) {
        int col = off >> 1;
        int par = off & 1;
        int c   = row * 2 + par;
        if (col < NCOL) {
            int p = col >> 1;
            int k = col & 1;
            #pragma unroll 8
            for (int o = 0; o < OCH; ++o)
                v += Wlin[k * OCH + o] * Wt[(size_t)o * (CCH * PP) + c * PP + p];
        }
    }
    Bp[idx] = v;
    if (idx < NC) {
        float s = blin[idx];
        #pragma unroll 8
        for (int o = 0; o < OCH; ++o)
            s += Wlin[idx * OCH + o] * bias[o];
        bfold[idx] = s;
    }
}

// ---------------------------------------------------------------------------
// Kernel 2: WMMA GEMM  Y[pix][col] = sum_c x[pix,c] * Wfold[c][col]
//   A: [16 pixels x 128 ch] fp32 per wave (global), B: paired fp32 in LDS
//   V_WMMA_F32_16X16X4_F32, 32 K-steps x 4 N-tiles per wave
// ---------------------------------------------------------------------------
__global__ void __launch_bounds__(256)
calib_gemm_kernel(const float* __restrict__ x,
                  const float* __restrict__ Bp,
                  float* __restrict__ Y) {
    __shared__ float Blds[BTOT];                    // 40 KB

    const int tid  = threadIdx.x;
    const int wave = tid >> 5;
    const int lane = tid & 31;

    // cooperative load of paired folded weights into LDS
    for (int i = tid; i < BTOT; i += 256)
        Blds[i] = Bp[i];
    __syncthreads();

    // 16 consecutive pixels per wave; 3136 % 16 == 0 so no batch crossing
    const int pixBase = (blockIdx.x * 8 + wave) * 16;

    // A-operand addressing per ISA 16x4 f32 layout:
    //   lanes 0-15 : M = lane,    K = {k0, k0+1}
    //   lanes16-31 : M = lane-16, K = {k0+2, k0+3}
    const int laneM = lane & 15;
    const int kHalf = (lane >> 4) << 1;             // 0 or 2
    const int pm    = pixBase + laneM;
    const int n     = pm / HW;
    const int hw    = pm - n * HW;
    const float* xb = x + (size_t)n * CCH * HW + hw;   // + c*HW per channel

    v8f acc0 = {}, acc1 = {}, acc2 = {}, acc3 = {};

    #pragma unroll 4
    for (int k0 = 0; k0 < CCH; k0 += 4) {
        const int c = k0 + kHalf;                   // even
        v2f a;
        a.x = xb[(size_t)c * HW];
        a.y = xb[(size_t)(c + 1) * HW];

        // one aligned b64 per tile, tiles at +128B immediates
        const float* brow = &Blds[(c >> 1) * BROW + laneM * 2];
        v2f b0 = *(const v2f*)(brow + 0);
        v2f b1 = *(const v2f*)(brow + 32);
        v2f b2 = *(const v2f*)(brow + 64);
        v2f b3 = *(const v2f*)(brow + 96);

        acc0 = __builtin_amdgcn_wmma_f32_16x16x4_f32(false, a, false, b0,
                                                     (short)0, acc0, false, false);
        acc1 = __builtin_amdgcn_wmma_f32_16x16x4_f32(false, a, false, b1,
                                                     (short)0, acc1, false, false);
        acc2 = __builtin_amdgcn_wmma_f32_16x16x4_f32(false, a, false, b2,
                                                     (short)0, acc2, false, false);
        acc3 = __builtin_amdgcn_wmma_f32_16x16x4_f32(false, a, false, b3,
                                                     (short)0, acc3, false, false);
    }

    // D layout: VGPR r -> (M=r, N=lane) for lanes 0-15, (M=r+8, N=lane-16) else
    const int Mofs = (lane >> 4) << 3;              // 0 or 8
    v8f accs[4] = {acc0, acc1, acc2, acc3};
    #pragma unroll
    for (int t = 0; t < 4; ++t) {
        const int N = laneM + t * 16;
        if (N < NCOL) {
            #pragma unroll
            for (int r = 0; r < 8; ++r) {
                const int M = r + Mofs;
                Y[(size_t)(pixBase + M) * NCOL + N] = accs[t][r];
            }
        }
    }
}

// ---------------------------------------------------------------------------
// Kernel 3: gather 25 shifted neighbors, drift epilogue -> out (N,H,W,2)
// ---------------------------------------------------------------------------
__global__ void calib_reduce_kernel(const float* __restrict__ Y,
                                    const float* __restrict__ bfold,
                                    float* __restrict__ out) {
    int pix = blockIdx.x * 256 + threadIdx.x;
    if (pix >= NPIX) return;
    int n  = pix / HW;
    int hw = pix - n * HW;
    int h  = hw / WW;
    int w  = hw - h * WW;

    const float b0 = bfold[0];
    const float b1 = bfold[1];

    float cm0[PP], cm1[PP];
    #pragma unroll
    for (int p = 0; p < PP; ++p) {
        int i   = p / KK, j = p - i * KK;
        int hh  = h + i - 2;
        int ww2 = w + j - 2;
        float v0 = 0.f, v1 = 0.f;
        if (hh >= 0 && hh < HH && ww2 >= 0 && ww2 < WW) {
            const float* yr = Y + (size_t)(n * HW + hh * WW + ww2) * NCOL + p * NC;
            v0 = yr[0];
            v1 = yr[1];
        }
        cm0[p] = v0 + b0;
        cm1[p] = v1 + b1;
    }

    #pragma unroll
    for (int k = 0; k < NC; ++k) {
        const float* cm = (k == 0) ? cm0 : cm1;
        float asum = 0.f, ys = 0.f, xs = 0.f, cs = 0.f;
        #pragma unroll
        for (int p = 0; p < PP; ++p) {
            float c  = cm[p];
            float a  = fabsf(c);
            float yo = (float)(p / KK - 2);
            float xo = (float)(p % KK - 2);
            asum += a;
            ys   += a * yo;
            xs   += a * xo;
            cs   += c;
        }
        float yd    = ys / asum;
        float xd    = xs / asum;
        float drift = sqrtf(xd * xd + yd * yd);
        out[(size_t)pix * NC + k] = cs * expf(-0.5f * drift);
    }
}

// ---------------------------------------------------------------------------
extern "C" void kernel_launch(void* const* d_in, const int* in_sizes, int n_in,
                              void* d_out, int out_size, void* d_ws, size_t ws_size,
                              hipStream_t stream) {
    const float* x    = (const float*)d_in[0];   // (8,128,56,56)
    const float* Wt   = (const float*)d_in[1];   // (64,128,5,5)
    const float* bias = (const float*)d_in[2];   // (64,)
    const float* Wlin = (const float*)d_in[3];   // (2,64)
    const float* blin = (const float*)d_in[4];   // (2,)
    float* out = (float*)d_out;                  // (8,56,56,2)

    float* wsf   = (float*)d_ws;
    float* Bp    = wsf;
    float* bfold = wsf + WS_BF;
    float* Y     = wsf + WS_Y;

    // 1) fold Wlin/blin into conv weights (paired, bank-swizzled layout)
    calib_fold_kernel<<<(BTOT + 255) / 256, 256, 0, stream>>>(
        Wt, bias, Wlin, blin, Bp, bfold);

    // 2) dense fp32 WMMA GEMM: Y[25088 x 50] = X[25088 x 128] * Wfold[128 x 50]
    calib_gemm_kernel<<<NPIX / 128, 256, 0, stream>>>(x, Bp, Y);

    // 3) shifted gather + drift epilogue
    calib_reduce_kernel<<<(NPIX + 255) / 256, 256, 0, stream>>>(Y, bfold, out);
}